// POPVAE_2765958938817
// MI455X (gfx1250) — compile-verified
//
#include <hip/hip_runtime.h>
#include <hip/hip_bf16.h>

// ---------------------------------------------------------------------------
// POPVAE forward for MI455X (gfx1250, wave32).
// GEMMs use native f32 WMMA (V_WMMA_F32_16X16X4_F32). Anchor grid_sample is an
// exact integer shift and is fused into the downstream bilinear samples, so no
// 67MB intermediates are materialized. Decoded parts are batch-invariant and
// staged in LDS.
// ---------------------------------------------------------------------------

#define HH 64
#define WW 64
#define PP 16
#define PHH 16
#define PWW 16
#define ZD 32
#define HD 1024
#define UD 3
#define BB 256

typedef __attribute__((ext_vector_type(2))) float v2f;
typedef __attribute__((ext_vector_type(8))) float v8f;

// ---------------- GEMM + bias + ReLU via f32 WMMA --------------------------
// C[M,N] = relu(A[M,K] @ W[K,N] + bias[N]); M from grid.y*16, block = 4 waves,
// each wave computes one 16x16 tile (block covers 16x64).
__global__ void gemm_relu_wmma(const float* __restrict__ A,
                               const float* __restrict__ W,
                               const float* __restrict__ bias,
                               float* __restrict__ C, int K, int N) {
    const int wave = threadIdx.x >> 5;
    const int lane = threadIdx.x & 31;
    const int l15  = lane & 15;
    const int hi   = lane >> 4;

    const int arow = blockIdx.y * 16 + l15;                 // A row for this lane
    const int col  = blockIdx.x * 64 + wave * 16 + l15;     // N index (B & C/D)
    const float* Arow = A + (size_t)arow * K;

    v8f acc = {0.f, 0.f, 0.f, 0.f, 0.f, 0.f, 0.f, 0.f};
    for (int k = 0; k < K; k += 4) {
        // A 16x4 f32: lane l -> M=l&15, K = 2*(l>>4) + {0,1}  (8B aligned)
        v2f a = *(const v2f*)(Arow + k + 2 * hi);
        // B 4x16 f32 (mirrored): lane l -> N=l&15, K = 2*(l>>4) + {0,1}
        v2f b;
        b.x = W[(size_t)(k + 2 * hi) * N + col];
        b.y = W[(size_t)(k + 2 * hi + 1) * N + col];
        acc = __builtin_amdgcn_wmma_f32_16x16x4_f32(false, a, false, b,
                                                    (short)0, acc, false, false);
    }

    // C/D layout: VGPR r -> row = r + 8*hi, col = l15
    const int crow0 = blockIdx.y * 16 + 8 * hi;
    const float bv = bias[col];
#pragma unroll
    for (int r = 0; r < 8; ++r) {
        float v = acc[r] + bv;
        C[(size_t)(crow0 + r) * N + col] = v > 0.f ? v : 0.f;
    }
}

// ---------------- heads: 102 dots of length 1024 per batch row -------------
__global__ void heads_kernel(const float* __restrict__ hid,
                             const float* __restrict__ Wum, const float* __restrict__ bum,
                             const float* __restrict__ Wuv, const float* __restrict__ buv,
                             const float* __restrict__ Wgm, const float* __restrict__ bgm,
                             const float* __restrict__ Wgv, const float* __restrict__ bgv,
                             float* __restrict__ umq, float* __restrict__ uvq,
                             float* __restrict__ gmq, float* __restrict__ gvq) {
    const int gw   = blockIdx.x * (blockDim.x >> 5) + (threadIdx.x >> 5);
    const int lane = threadIdx.x & 31;
    if (gw >= BB * 102) return;
    const int b = gw / 102;
    const int j = gw % 102;

    const float* Wm; const float* bb; int nd; int jj; float* dst;
    if (j < 48)      { Wm = Wum; bb = bum; nd = PP * UD; jj = j;      dst = umq + b * 48 + jj; }
    else if (j < 96) { Wm = Wuv; bb = buv; nd = PP * UD; jj = j - 48; dst = uvq + b * 48 + jj; }
    else if (j < 99) { Wm = Wgm; bb = bgm; nd = UD;      jj = j - 96; dst = gmq + b * UD + jj; }
    else             { Wm = Wgv; bb = bgv; nd = UD;      jj = j - 99; dst = gvq + b * UD + jj; }

    const float* h = hid + (size_t)b * HD;
    float s = 0.f;
    for (int t = lane; t < HD; t += 32) s += h[t] * Wm[(size_t)t * nd + jj];
    for (int off = 16; off > 0; off >>= 1) s += __shfl_xor(s, off, 32);
    if (lane == 0) *dst = s + bb[jj];
}

// ---------------- per-part transform params (cos,sin,tx,ty) ----------------
__global__ void u_params_kernel(const float* __restrict__ umq, const float* __restrict__ uvq,
                                const float* __restrict__ eps_u, float* __restrict__ uparams) {
    const int bp = blockIdx.x * blockDim.x + threadIdx.x;
    if (bp >= BB * PP) return;
    const int b = bp >> 4, p = bp & 15;
    float u[3];
#pragma unroll
    for (int d = 0; d < 3; ++d) {
        const int o = b * 48 + p * 3 + d;
        float mu  = tanhf(umq[o]);
        float pre = uvq[o];
        float var = expf(pre > -6.f ? pre : -6.f);   // F.threshold(x,-6,-6)
        u[d] = mu + var * eps_u[o];
    }
    uparams[bp * 4 + 0] = cosf(u[0]);
    uparams[bp * 4 + 1] = sinf(u[0]);
    uparams[bp * 4 + 2] = u[1];
    uparams[bp * 4 + 3] = u[2];
}

__global__ void g_params_kernel(const float* __restrict__ gmq, const float* __restrict__ gvq,
                                const float* __restrict__ eps_g, float* __restrict__ gparams) {
    const int b = blockIdx.x * blockDim.x + threadIdx.x;
    if (b >= BB) return;
    float g[3];
#pragma unroll
    for (int d = 0; d < 3; ++d) {
        const int o = b * 3 + d;
        float mu  = tanhf(gmq[o]);
        float pre = gvq[o];
        float var = expf(pre > -6.f ? pre : 6.f);    // F.threshold(x,-6,6)
        g[d] = mu + var * eps_g[o];
    }
    gparams[b * 4 + 0] = cosf(g[0]);
    gparams[b * 4 + 1] = sinf(g[0]);
    gparams[b * 4 + 2] = g[1];
    gparams[b * 4 + 3] = g[2];
}

// ---------------- decoded parts: sigmoid(colsum(Wz)), batch-invariant ------
__global__ void parts_kernel(const float* __restrict__ Wz, float* __restrict__ parts0) {
    const int j = blockIdx.x * blockDim.x + threadIdx.x;
    if (j >= PP * PHH * PWW) return;
    float s = 0.f;
#pragma unroll 4
    for (int r = 0; r < ZD; ++r) s += Wz[(size_t)r * (PP * PHH * PWW) + j];
    parts0[j] = 1.f / (1.f + expf(-s));
}

// ---------------- fused sampling helpers -----------------------------------
// Tap into "anchored" parts image: outer-sample tap (xi,yi) must be in [0,64)
// AND the anchor-shifted source (xi+sx, yi+sy) must land in the padded 16x16
// part region [24,40) (zero padding everywhere else).
__device__ __forceinline__ float part_tap(const float* sParts, int p, int xi, int yi,
                                          int sx, int sy) {
    if ((unsigned)xi >= 64u || (unsigned)yi >= 64u) return 0.f;
    const int xs = xi + sx, ys = yi + sy;
    if (xs < 24 || xs >= 40 || ys < 24 || ys >= 40) return 0.f;
    return sParts[p * 256 + (ys - 24) * 16 + (xs - 24)];
}

__device__ __forceinline__ float km_tap(const float* __restrict__ Mbp, int xi, int yi,
                                        int sx, int sy) {
    if ((unsigned)xi >= 64u || (unsigned)yi >= 64u) return 0.f;
    const int xs = xi + sx, ys = yi + sy;
    if ((unsigned)xs >= 64u || (unsigned)ys >= 64u) return 0.f;
    return Mbp[ys * 64 + xs];
}

__device__ __forceinline__ float img_tap(const float* __restrict__ img, int xi, int yi) {
    if ((unsigned)xi >= 64u || (unsigned)yi >= 64u) return 0.f;
    return img[yi * 64 + xi];
}

__device__ __forceinline__ float bilin_part(const float* sParts, int p,
                                            float px, float py, int sx, int sy) {
    const float fx = floorf(px), fy = floorf(py);
    const int x0 = (int)fx, y0 = (int)fy;
    const float wx = px - fx, wy = py - fy;
    return part_tap(sParts, p, x0,     y0,     sx, sy) * (1.f - wx) * (1.f - wy)
         + part_tap(sParts, p, x0 + 1, y0,     sx, sy) * wx         * (1.f - wy)
         + part_tap(sParts, p, x0,     y0 + 1, sx, sy) * (1.f - wx) * wy
         + part_tap(sParts, p, x0 + 1, y0 + 1, sx, sy) * wx         * wy;
}

__device__ __forceinline__ float bilin_km(const float* __restrict__ Mbp,
                                          float px, float py, int sx, int sy) {
    const float fx = floorf(px), fy = floorf(py);
    const int x0 = (int)fx, y0 = (int)fy;
    const float wx = px - fx, wy = py - fy;
    return km_tap(Mbp, x0,     y0,     sx, sy) * (1.f - wx) * (1.f - wy)
         + km_tap(Mbp, x0 + 1, y0,     sx, sy) * wx         * (1.f - wy)
         + km_tap(Mbp, x0,     y0 + 1, sx, sy) * (1.f - wx) * wy
         + km_tap(Mbp, x0 + 1, y0 + 1, sx, sy) * wx         * wy;
}

__device__ __forceinline__ float bilin_img(const float* __restrict__ img,
                                           float px, float py) {
    const float fx = floorf(px), fy = floorf(py);
    const int x0 = (int)fx, y0 = (int)fy;
    const float wx = px - fx, wy = py - fy;
    return img_tap(img, x0,     y0)     * (1.f - wx) * (1.f - wy)
         + img_tap(img, x0 + 1, y0)     * wx         * (1.f - wy)
         + img_tap(img, x0,     y0 + 1) * (1.f - wx) * wy
         + img_tap(img, x0 + 1, y0 + 1) * wx         * wy;
}

// ---------------- composite: fused anchor + per-part warp + blend ----------
// grid = (16 pixel-chunks, B), block = 256 threads (one pixel each).
__global__ void composite_kernel(const float* __restrict__ parts0,
                                 const float* __restrict__ uparams,
                                 const float* __restrict__ M,
                                 float* __restrict__ xcomp) {
    __shared__ float sParts[PP * PHH * PWW];   // 16 KB
    __shared__ float sU[PP * 4];
    const int b   = blockIdx.y;
    const int tid = threadIdx.x;
    for (int i = tid; i < PP * PHH * PWW; i += 256) sParts[i] = parts0[i];
    if (tid < PP * 4) sU[tid] = uparams[b * PP * 4 + tid];
    __syncthreads();

    const int idx = blockIdx.x * 256 + tid;
    const int y = idx >> 6, x = idx & 63;
    const float xn = (2.f * x + 1.f) / 64.f - 1.f;
    const float yn = (2.f * y + 1.f) / 64.f - 1.f;
    const float* __restrict__ Mb = M + (size_t)b * PP * HH * WW;

    float num = 0.f, den = 0.f;
#pragma unroll
    for (int p = 0; p < PP; ++p) {
        const float c = sU[p * 4 + 0], s = sU[p * 4 + 1];
        const float tx = sU[p * 4 + 2], ty = sU[p * 4 + 3];
        const int sx = 8 * (p & 3) - 12;   // anchor shift = tx_a * 32
        const int sy = 8 * (p >> 2) - 12;

        // per-part rotation+translation sample of anchored parts
        const float px = (c * xn - s * yn + tx + 1.f) * 32.f - 0.5f;
        const float py = (s * xn + c * yn + ty + 1.f) * 32.f - 0.5f;
        const float xpv = bilin_part(sParts, p, px, py, sx, sy);

        // translation-only sample of anchored kernel mask
        const float pxK = (float)x + tx * 32.f;
        const float pyK = (float)y + ty * 32.f;
        const float kmv = bilin_km(Mb + (size_t)p * HH * WW, pxK, pyK, sx, sy);

        num += xpv * kmv;
        den += kmv;
    }
    if (den == 0.f) den = 1.f;
    xcomp[(size_t)b * (HH * WW) + idx] = num / den;
}

// ---------------- global rot+trans resample --------------------------------
__global__ void global_kernel(const float* __restrict__ xcomp,
                              const float* __restrict__ gparams,
                              float* __restrict__ out) {
    const int b   = blockIdx.y;
    const int idx = blockIdx.x * 256 + threadIdx.x;
    const int y = idx >> 6, x = idx & 63;
    const float c = gparams[b * 4 + 0], s = gparams[b * 4 + 1];
    const float tx = gparams[b * 4 + 2], ty = gparams[b * 4 + 3];
    const float xn = (2.f * x + 1.f) / 64.f - 1.f;
    const float yn = (2.f * y + 1.f) / 64.f - 1.f;
    const float px = (c * xn - s * yn + tx + 1.f) * 32.f - 0.5f;
    const float py = (s * xn + c * yn + ty + 1.f) * 32.f - 0.5f;
    out[(size_t)b * (HH * WW) + idx] = bilin_img(xcomp + (size_t)b * (HH * WW), px, py);
}

// ---------------------------------------------------------------------------
extern "C" void kernel_launch(void* const* d_in, const int* in_sizes, int n_in,
                              void* d_out, int out_size, void* d_ws, size_t ws_size,
                              hipStream_t stream) {
    const float* inputs = (const float*)d_in[0];
    const float* W1   = (const float*)d_in[1];
    const float* b1   = (const float*)d_in[2];
    const float* W2   = (const float*)d_in[3];
    const float* b2   = (const float*)d_in[4];
    const float* Wum  = (const float*)d_in[5];
    const float* bum  = (const float*)d_in[6];
    const float* Wuv  = (const float*)d_in[7];
    const float* buv  = (const float*)d_in[8];
    const float* Wgm  = (const float*)d_in[9];
    const float* bgm  = (const float*)d_in[10];
    const float* Wgv  = (const float*)d_in[11];
    const float* bgv  = (const float*)d_in[12];
    const float* Wz   = (const float*)d_in[13];
    const float* M    = (const float*)d_in[14];
    const float* eps_u = (const float*)d_in[15];
    const float* eps_g = (const float*)d_in[16];
    float* out = (float*)d_out;

    // workspace layout (floats)
    float* ws = (float*)d_ws;
    float* hid1    = ws;                   // 262144
    float* hid2    = hid1 + 262144;        // 262144
    float* umq     = hid2 + 262144;        // 12288
    float* uvq     = umq + 12288;          // 12288
    float* gmq     = uvq + 12288;          // 768
    float* gvq     = gmq + 768;            // 768
    float* uparams = gvq + 768;            // 16384
    float* gparams = uparams + 16384;      // 1024
    float* parts0  = gparams + 1024;       // 4096
    float* xcomp   = parts0 + 4096;        // 1048576
    // total ~1.62M floats (~6.5 MB)

    // hid1 = relu(inputs @ W1 + b1): M=256, K=4096, N=1024
    gemm_relu_wmma<<<dim3(HD / 64, BB / 16), 128, 0, stream>>>(inputs, W1, b1, hid1, HH * WW, HD);
    // hid2 = relu(hid1 @ W2 + b2): M=256, K=1024, N=1024
    gemm_relu_wmma<<<dim3(HD / 64, BB / 16), 128, 0, stream>>>(hid1, W2, b2, hid2, HD, HD);

    // heads: 256 * 102 wave-dots, 8 waves per 256-thread block
    heads_kernel<<<(BB * 102) / 8, 256, 0, stream>>>(hid2, Wum, bum, Wuv, buv,
                                                     Wgm, bgm, Wgv, bgv,
                                                     umq, uvq, gmq, gvq);

    u_params_kernel<<<(BB * PP) / 256, 256, 0, stream>>>(umq, uvq, eps_u, uparams);
    g_params_kernel<<<1, 256, 0, stream>>>(gmq, gvq, eps_g, gparams);
    parts_kernel<<<(PP * PHH * PWW) / 256, 256, 0, stream>>>(Wz, parts0);

    composite_kernel<<<dim3(16, BB), 256, 0, stream>>>(parts0, uparams, M, xcomp);
    global_kernel<<<dim3(16, BB), 256, 0, stream>>>(xcomp, gparams, out);
}